// HungarianMatcher_79688823209989
// MI455X (gfx1250) — compile-verified
//
#include <hip/hip_runtime.h>
#include <cstddef>

// Problem constants (from setup_inputs)
#define B_ 64
#define Q_ 900
#define T_ 256
#define G_ 32
#define QT_ 57          // ceil(900/16)
#define EPS_ 1e-6f

typedef float v2f __attribute__((ext_vector_type(2)));
typedef float v8f __attribute__((ext_vector_type(8)));

// ---------------------------------------------------------------------------
// Phase A: prob[b,q] = sigmoid(max_t logits[b,q,t]).  One wave per (b,q) row:
// 32 lanes x 2 float4 = 256 contiguous floats, then 5-step shfl_xor max.
// Bandwidth-bound (59 MB read) -> coalesced b128 loads + global_prefetch.
// ---------------------------------------------------------------------------
__global__ __launch_bounds__(256) void maxprob_kernel(const float* __restrict__ logits,
                                                      float* __restrict__ prob) {
    int wave = threadIdx.x >> 5;
    int lane = threadIdx.x & 31;
    int q = blockIdx.x * 8 + wave;            // flat (b,q) in [0, B*Q)
    if (q >= B_ * Q_) return;

    const float* base = logits + (size_t)q * T_;
    if (q + 8 < B_ * Q_)                       // gfx1250 global_prefetch_b8 path
        __builtin_prefetch(base + 8 * T_ + lane * 8, 0, 1);

    const float4* lg = reinterpret_cast<const float4*>(base);
    float4 a = lg[lane];                       // floats [lane*4 .. lane*4+3]
    float4 b = lg[32 + lane];                  // floats [128+lane*4 ..]
    float m = fmaxf(fmaxf(fmaxf(a.x, a.y), fmaxf(a.z, a.w)),
                    fmaxf(fmaxf(b.x, b.y), fmaxf(b.z, b.w)));
#pragma unroll
    for (int off = 16; off > 0; off >>= 1)
        m = fmaxf(m, __shfl_xor(m, off, 32));

    if (lane == 0) {
        // numerically stable sigmoid (monotone => max(sigmoid) == sigmoid(max))
        float s = (m >= 0.0f) ? 1.0f / (1.0f + expf(-m))
                              : expf(m) / (1.0f + expf(m));
        prob[q] = s;
    }
}

// ---------------------------------------------------------------------------
// Phase B: C[b,q,g] = -prob + 5*L1(pred,tgt) + 2*(-giou).
// One wave computes a 16(q) x 16(g) tile laid out in the ISA C/D layout:
//   VGPR r, lane l -> M = r + 8*(l>=16), N = l&15.
// The rank-1 class-cost term is applied with V_WMMA_F32_16X16X4_F32:
//   A[M][0] = -prob[q0+M], B[0][N] = 1  =>  D = C - prob (exact fp32).
// ---------------------------------------------------------------------------
__global__ __launch_bounds__(128) void cost_kernel(const float* __restrict__ pred_boxes,
                                                   const float* __restrict__ tgt_boxes,
                                                   const float* __restrict__ prob,
                                                   float* __restrict__ C) {
    int lane = threadIdx.x & 31;
    int tile = blockIdx.x * 4 + (threadIdx.x >> 5);
    if (tile >= B_ * QT_ * 2) return;          // uniform per wave; EXEC stays full
    int gt = tile & 1;
    int qt = (tile >> 1) % QT_;
    int b  = tile / (2 * QT_);
    int q0 = qt * 16;
    int g  = gt * 16 + (lane & 15);
    int half = lane >> 4;                      // 0 or 1

    const float4* tb4 = reinterpret_cast<const float4*>(tgt_boxes);
    float4 t = tb4[b * G_ + g];
    float tx0 = fminf(fmaxf(t.x - 0.5f * t.z, 0.0f), 1.0f);
    float ty0 = fminf(fmaxf(t.y - 0.5f * t.w, 0.0f), 1.0f);
    float tx1 = fminf(fmaxf(t.x + 0.5f * t.z, 0.0f), 1.0f);
    float ty1 = fminf(fmaxf(t.y + 0.5f * t.w, 0.0f), 1.0f);
    float area2 = (tx1 - tx0) * (ty1 - ty0);

    const float4* pb4 = reinterpret_cast<const float4*>(pred_boxes);

    v8f acc;
#pragma unroll
    for (int r = 0; r < 8; ++r) {
        int q  = q0 + r + 8 * half;            // row M = r + 8*half
        int qc = (q < Q_) ? q : (Q_ - 1);
        float4 p = pb4[b * Q_ + qc];
        // L1 cdist on raw cxcywh
        float cb = fabsf(p.x - t.x) + fabsf(p.y - t.y) +
                   fabsf(p.z - t.z) + fabsf(p.w - t.w);
        // xyxy + clip
        float px0 = fminf(fmaxf(p.x - 0.5f * p.z, 0.0f), 1.0f);
        float py0 = fminf(fmaxf(p.y - 0.5f * p.w, 0.0f), 1.0f);
        float px1 = fminf(fmaxf(p.x + 0.5f * p.z, 0.0f), 1.0f);
        float py1 = fminf(fmaxf(p.y + 0.5f * p.w, 0.0f), 1.0f);
        float area1 = (px1 - px0) * (py1 - py0);
        float ltx = fmaxf(px0, tx0), lty = fmaxf(py0, ty0);
        float rbx = fminf(px1, tx1), rby = fminf(py1, ty1);
        float inter = fmaxf(rbx - ltx, 0.0f) * fmaxf(rby - lty, 0.0f);
        float a_sum = area1 + area2;
        float uni   = a_sum - inter;
        float iou   = inter / (uni + EPS_);
        float cx0 = fminf(px0, tx0), cy0 = fminf(py0, ty0);
        float cx1 = fmaxf(px1, tx1), cy1 = fmaxf(py1, ty1);
        float area_c = fmaxf(cx1 - cx0, 0.0f) * fmaxf(cy1 - cy0, 0.0f);
        float union2 = a_sum - iou * a_sum / (iou + EPS_);   // as written in source
        float giou   = iou - (area_c - union2) / (area_c + EPS_);
        acc[r] = 5.0f * cb - 2.0f * giou;      // COST_BBOX*bbox + COST_GIOU*(-giou)
    }

    // Rank-1 class cost via the matrix pipe: D = A(16x4) * B(4x16) + C.
    // A layout: lanes 0-15 hold M=lane, a.x=K0, a.y=K1; lanes 16-31 hold K2/K3.
    v2f A  = {0.0f, 0.0f};
    v2f Bm = {0.0f, 0.0f};
    if (lane < 16) {
        int q = q0 + lane;
        A.x  = (q < Q_) ? -prob[b * Q_ + q] : 0.0f;  // A[M][0]
        Bm.x = 1.0f;                                  // B[0][N]
    }
    acc = __builtin_amdgcn_wmma_f32_16x16x4_f32(
        /*neg_a=*/false, A, /*neg_b=*/false, Bm,
        /*c_mod=*/(short)0, acc, /*reuse_a=*/false, /*reuse_b=*/false);

    float* Cb = C + (size_t)b * Q_ * G_;
#pragma unroll
    for (int r = 0; r < 8; ++r) {
        int q = q0 + r + 8 * half;
        if (q < Q_) Cb[(size_t)q * G_ + g] = acc[r];   // 64B contiguous per half-wave
    }
}

// ---------------------------------------------------------------------------
// Phase C: Jonker-Volgenant shortest augmenting path, one wave32 per batch.
// cost(i=g, j=q) = C[b][q][g]; n=32 rows, m=900 cols. State in LDS, column
// scans strided over 32 lanes, argmin via shfl_xor with index tie-break
// (matches np.argmin first-of-ties). Arithmetic in double (matches float64).
// ---------------------------------------------------------------------------
__global__ __launch_bounds__(32) void lsa_kernel(const float* __restrict__ C,
                                                 float* __restrict__ pred_idx,
                                                 float* __restrict__ gt_idx) {
    const int n = G_, m = Q_;
    const double INF = 1e18;
    int b = blockIdx.x;
    int t = threadIdx.x;
    const float* Cb = C + (size_t)b * Q_ * G_;

    __shared__ double u[G_];
    __shared__ double v[Q_ + 1];
    __shared__ double minv[Q_];
    __shared__ int    way[Q_];
    __shared__ int    p[Q_ + 1];               // column -> row (col m virtual)
    __shared__ unsigned char used[Q_ + 1];
    __shared__ int    col4row[G_];

    for (int j = t; j <= m; j += 32) { v[j] = 0.0; p[j] = -1; }
    u[t] = 0.0;
    __syncthreads();

    for (int i = 0; i < n; ++i) {
        if (t == 0) { p[m] = i; used[m] = 0; }
        for (int j = t; j < m; j += 32) { minv[j] = INF; way[j] = 0; used[j] = 0; }
        __syncthreads();

        int j0 = m;
        while (true) {                          // numpy: while p[j0] != -1
            if (t == 0) used[j0] = 1;
            __syncthreads();
            int i0 = p[j0];
            double ui0 = u[i0];

            double lmin = INF; int lidx = m;
            for (int j = t; j < m; j += 32) {
                if (!used[j]) {
                    double cur = (double)Cb[(size_t)j * G_ + i0] - ui0 - v[j];
                    if (cur < minv[j]) { minv[j] = cur; way[j] = j0; }
                    double mv = minv[j];
                    if (mv < lmin) { lmin = mv; lidx = j; }  // keep smallest j on ties
                }
            }
#pragma unroll
            for (int off = 16; off > 0; off >>= 1) {
                double ov = __shfl_xor(lmin, off, 32);
                int    oi = __shfl_xor(lidx, off, 32);
                if (ov < lmin || (ov == lmin && oi < lidx)) { lmin = ov; lidx = oi; }
            }
            int j1 = lidx; double delta = lmin;

            // u[p[used]] += delta; v[used] -= delta; minv[free] -= delta
            for (int j = t; j <= m; j += 32) {
                if (used[j]) { u[p[j]] += delta; v[j] -= delta; }
                else if (j < m) minv[j] -= delta;
            }
            __syncthreads();
            j0 = j1;
            if (p[j0] == -1) break;
        }
        if (t == 0) {                           // augment
            int j = j0;
            while (j != m) { int jn = way[j]; p[j] = p[jn]; j = jn; }
        }
        __syncthreads();
    }

    for (int j = t; j < m; j += 32) { int r = p[j]; if (r >= 0) col4row[r] = j; }
    __syncthreads();

    // argsort(col4row) via parallel ranking (values are distinct)
    int cj = col4row[t];
    int rank = 0;
    for (int k = 0; k < n; ++k) rank += (col4row[k] < cj) ? 1 : 0;
    pred_idx[b * G_ + rank] = (float)cj;        // col4row[order]
    gt_idx[b * G_ + rank]   = (float)t;         // order
}

// ---------------------------------------------------------------------------
extern "C" void kernel_launch(void* const* d_in, const int* in_sizes, int n_in,
                              void* d_out, int out_size, void* d_ws, size_t ws_size,
                              hipStream_t stream) {
    const float* pred_logits = (const float*)d_in[0];  // [B,Q,T]
    const float* pred_boxes  = (const float*)d_in[1];  // [B,Q,4]
    const float* tgt_boxes   = (const float*)d_in[2];  // [B,G,4]

    float* out      = (float*)d_out;
    float* C        = out;                                   // [B,Q,G]
    float* pred_idx = out + (size_t)B_ * Q_ * G_;            // [B,G]
    float* gt_idx   = pred_idx + (size_t)B_ * G_;            // [B,G]
    float* prob     = (float*)d_ws;                          // [B*Q] scratch

    // A: 57600 rows, 8 waves/block -> 7200 blocks
    maxprob_kernel<<<(B_ * Q_) / 8, 256, 0, stream>>>(pred_logits, prob);

    // B: 64 * 57 * 2 = 7296 tiles, 4 waves/block -> 1824 blocks (exact)
    int tiles = B_ * QT_ * 2;
    cost_kernel<<<(tiles + 3) / 4, 128, 0, stream>>>(pred_boxes, tgt_boxes, prob, C);

    // C: one wave32 per batch
    lsa_kernel<<<B_, 32, 0, stream>>>(C, pred_idx, gt_idx);
}